// DoublePSMCosineModule_6201932775920
// MI455X (gfx1250) — compile-verified
//
#include <hip/hip_runtime.h>
#include <hip/hip_bf16.h>

// Problem constants (from reference): B=4, C=512, H=96, W=320, DEPTH_CH=48
#define BB   4
#define CC   512
#define HH   96
#define WW   320
#define DCH  48
#define CHW  (CC * HH * WW)     // per-batch elements
#define HWP  (HH * WW)          // c-stride in elements (30720)
#define NTI  (WW / 16)          // 20 w-tiles

typedef __attribute__((ext_vector_type(2))) float v2f;
typedef __attribute__((ext_vector_type(8))) float v8f;

// ---------------------------------------------------------------------------
// Pass 1: materialize right_shifted (bilinear grid_sample with transposed
// coords, zero padding) into workspace.  Pure memory-bound gather.
//   xpix(h) = h*W/H - 0.5
//   ypix(w) = w*H/W - H/(2C) - 0.5
// ---------------------------------------------------------------------------
__global__ __launch_bounds__(256) void shift_kernel(const float* __restrict__ right,
                                                    float* __restrict__ outp) {
    int idx = blockIdx.x * 256 + threadIdx.x;
    const int total = BB * CC * HH * WW;
    if (idx >= total) return;

    int w  = idx % WW;
    int h  = (idx / WW) % HH;
    int bc = idx / (WW * HH);            // b*C + c

    float xpix = (float)h * ((float)WW / (float)HH) - 0.5f;
    float ypix = (float)w * ((float)HH / (float)WW)
                 - ((float)HH / (2.0f * (float)CC)) - 0.5f;

    float x0f = floorf(xpix), y0f = floorf(ypix);
    float wx1 = xpix - x0f,   wy1 = ypix - y0f;
    float wx0 = 1.0f - wx1,   wy0 = 1.0f - wy1;
    int xi = (int)x0f, yi = (int)y0f;

    const float* img = right + (size_t)bc * HWP;

    float v00 = 0.f, v01 = 0.f, v10 = 0.f, v11 = 0.f;
    bool vx0 = (xi >= 0) && (xi < WW);
    bool vx1 = (xi + 1 >= 0) && (xi + 1 < WW);
    bool vy0 = (yi >= 0) && (yi < HH);
    bool vy1 = (yi + 1 >= 0) && (yi + 1 < HH);
    int xc0 = min(max(xi, 0), WW - 1);
    int xc1 = min(max(xi + 1, 0), WW - 1);
    int yc0 = min(max(yi, 0), HH - 1);
    int yc1 = min(max(yi + 1, 0), HH - 1);
    if (vy0 && vx0) v00 = img[yc0 * WW + xc0];
    if (vy0 && vx1) v01 = img[yc0 * WW + xc1];
    if (vy1 && vx0) v10 = img[yc1 * WW + xc0];
    if (vy1 && vx1) v11 = img[yc1 * WW + xc1];

    outp[idx] = v00 * (wy0 * wx0) + v01 * (wy0 * wx1)
              + v10 * (wy1 * wx0) + v11 * (wy1 * wx1);
}

// ---------------------------------------------------------------------------
// Pass 2: banded matmul cost volume via V_WMMA_F32_16X16X4_F32.
// One block (4 wave32) per (b, h, source, w-tile i).  Wave t computes the
// 16x16 tile  M = L^T[16w x 512] * R[512 x 16w'] with j = i-3+t, then
// scatters the diagonal band d = w - w' in [0,48) into LDS and stores.
// ---------------------------------------------------------------------------
__global__ __launch_bounds__(128) void cost_kernel(const float* __restrict__ left,
                                                   const float* __restrict__ right,
                                                   const float* __restrict__ rshift,
                                                   float* __restrict__ out) {
    __shared__ float ldsL[CC * 16];     // L tile, [c][w_local], 32 KB
    __shared__ float ldsOut[DCH * 16];  // output staging [d][w_local], 3 KB

    const int tid = threadIdx.x;
    int bi = blockIdx.x;
    const int i   = bi % NTI;  bi /= NTI;     // w-tile index
    const int src = bi & 1;    bi >>= 1;      // 0: right, 1: right_shifted
    const int h   = bi % HH;
    const int b   = bi / HH;
    const int wbase = i * 16;

    const size_t base_bh = ((size_t)b * CC * HH + (size_t)h) * WW; // (b, c=0, h, 0)
    const float* Lp = left + base_bh;
    const float* Rp = (src ? rshift : right) + base_bh;

    // --- cooperative load of L tile into LDS: 512 rows x 16 floats ----------
    // 2048 float4 segments; thread t does 16 of them.
    for (int fi = tid; fi < CC * 4; fi += 128) {
        const int c = fi >> 2;
        const int seg = (fi & 3) << 2;
        float4 v = *(const float4*)(Lp + (size_t)c * HWP + wbase + seg);
        *(float4*)(&ldsL[c * 16 + seg]) = v;
    }
    // zero output staging
    for (int o = tid; o < DCH * 16; o += 128) ldsOut[o] = 0.0f;
    __syncthreads();

    const int wave = tid >> 5;
    const int lane = tid & 31;
    const int j = i - 3 + wave;            // w'-tile for this wave

    if (j >= 0) {
        const int n    = lane & 15;            // N (w') col / A row m
        const int toff = (lane >> 4) << 1;     // 0 for lanes 0-15, 2 for 16-31

        // B stream pointer: R[c = toff, h, 16j + n]
        const float* rp = Rp + (size_t)toff * HWP + (size_t)(16 * j + n);
        const float* lp = &ldsL[toff * 16 + n];

        v8f acc = {};
        #pragma unroll 4
        for (int kc = 0; kc < CC; kc += 4) {
            v2f a, bm;
            a.x  = lp[0];                      // A[m=n, k=toff]
            a.y  = lp[16];                     // A[m=n, k=toff+1]
            bm.x = rp[0];                      // B[k=toff,   n]
            bm.y = rp[HWP];                    // B[k=toff+1, n]
            acc = __builtin_amdgcn_wmma_f32_16x16x4_f32(
                /*neg_a=*/false, a, /*neg_b=*/false, bm,
                /*c_mod=*/(short)0, acc, /*reuse_a=*/false, /*reuse_b=*/false);
            lp += 4 * 16;
            rp += (size_t)4 * HWP;
        }

        // scatter 16x16 tile into band staging: d = 16(i-j) + w_local - n
        const int dbase = 16 * (i - j);
        const int mhi = (lane >> 4) << 3;      // +8 rows for lanes 16-31
        #pragma unroll
        for (int r = 0; r < 8; ++r) {
            const int wl = r + mhi;            // w_local = M
            const int d  = dbase + wl - n;
            if (d >= 0 && d < DCH)
                ldsOut[d * 16 + wl] = acc[r] * (1.0f / (float)CC);
        }
    }
    __syncthreads();

    // --- coalesced store: out[b, src*48 + d, h, wbase + wl] -----------------
    float* ob = out + (((size_t)(b * (2 * DCH) + src * DCH)) * HH + (size_t)h) * WW + wbase;
    for (int o = tid; o < DCH * 16; o += 128) {
        const int d  = o >> 4;
        const int wl = o & 15;
        ob[(size_t)d * HWP + wl] = ldsOut[o];
    }
}

extern "C" void kernel_launch(void* const* d_in, const int* in_sizes, int n_in,
                              void* d_out, int out_size, void* d_ws, size_t ws_size,
                              hipStream_t stream) {
    const float* left  = (const float*)d_in[0];
    const float* right = (const float*)d_in[1];
    float* out = (float*)d_out;
    float* rshift = (float*)d_ws;   // needs B*C*H*W*4 = 251,658,240 bytes

    // Pass 1: build right_shifted in workspace
    {
        const int total = BB * CC * HH * WW;
        const int blocks = (total + 255) / 256;
        shift_kernel<<<blocks, 256, 0, stream>>>(right, rshift);
    }
    // Pass 2: banded WMMA cost volume for both sources
    {
        const int blocks = BB * HH * 2 * NTI;   // 15360
        cost_kernel<<<blocks, 128, 0, stream>>>(left, right, rshift, out);
    }
}